// GraphNorm_78048145702990
// MI455X (gfx1250) — compile-verified
//
#include <hip/hip_runtime.h>

#define FEAT 256
#define NGRAPH 1024
#define EPS 1e-5f

typedef float v2f __attribute__((ext_vector_type(2)));
typedef float v8f __attribute__((ext_vector_type(8)));
typedef float f4  __attribute__((ext_vector_type(4)));

// ---------------------------------------------------------------------------
// Kernel 1: segment start offsets via binary search over sorted seg_ids.
// starts[g] = lower_bound(seg_ids, g), g in [0, NGRAPH]; starts[NGRAPH] = N.
// ---------------------------------------------------------------------------
__global__ void seg_starts_kernel(const int* __restrict__ seg_ids,
                                  int* __restrict__ starts, int n_nodes) {
    int g = blockIdx.x * blockDim.x + threadIdx.x;
    if (g > NGRAPH) return;
    int lo = 0, hi = n_nodes;
    while (lo < hi) {
        int mid = (lo + hi) >> 1;
        if (seg_ids[mid] < g) lo = mid + 1; else hi = mid;
    }
    starts[g] = lo;
}

// ---------------------------------------------------------------------------
// Kernel 2: one workgroup per segment. 512 threads = 16 waves.
// Pass A: per-feature sum / sumsq via V_WMMA_F32_16X16X4_F32 (A = x tile,
//         B = ones -> D[m,*] += row-sum of 4 nodes). Branch-free main loop,
//         2-chunk software pipeline; guarded tail chunk.
// Pass B: streaming normalize with b128 loads + non-temporal b128 stores.
// ---------------------------------------------------------------------------
__global__ __launch_bounds__(512)
void graphnorm_kernel(const float* __restrict__ x,
                      const float* __restrict__ gamma,
                      const float* __restrict__ beta,
                      const int*   __restrict__ starts,
                      float*       __restrict__ out) {
    __shared__ float lds_a[FEAT];   // sum   -> mean
    __shared__ float lds_b[FEAT];   // sumsq -> 1/(std+eps)

    const int g = blockIdx.x;
    const int s = starts[g];
    const int e = starts[g + 1];
    const int c = e - s;
    if (c == 0) return;

    const int tid  = threadIdx.x;
    const int wave = tid >> 5;           // 0..15 : feature tile
    const int lane = tid & 31;
    const int fl   = lane & 15;          // M within tile
    const int half = lane >> 4;          // selects K = {0,1} vs {2,3}
    const int f0   = wave << 4;
    const int feat = f0 + fl;

    // ---------------- Pass A: stats via WMMA ----------------
    v8f csum = {};
    v8f csq  = {};
    v2f bones; bones.x = 1.0f; bones.y = 1.0f;
    const int rx = 2 * half;             // A.x holds K = rx, A.y holds K = rx+1

    const int nfull = c >> 2;            // branch-free 4-row chunks
    // lane's pointer: A.x element of chunk 0 (row s+rx); A.y is +FEAT floats
    const float* px = x + (size_t)(s + rx) * FEAT + feat;

    int it = 0;
    // 2-chunk pipeline: 4 loads in flight before the 4 WMMAs
    for (; it + 1 < nfull; it += 2) {
        v2f a0, a1;
        a0.x = px[0];
        a0.y = px[FEAT];
        a1.x = px[4 * FEAT];
        a1.y = px[5 * FEAT];
        __builtin_prefetch(px + 8 * FEAT, 0, 3);
        px += 8 * FEAT;
        v2f q0 = a0 * a0;
        v2f q1 = a1 * a1;
        csum = __builtin_amdgcn_wmma_f32_16x16x4_f32(false, a0, false, bones,
                                                     (short)0, csum, false, false);
        csq  = __builtin_amdgcn_wmma_f32_16x16x4_f32(false, q0, false, bones,
                                                     (short)0, csq,  false, false);
        csum = __builtin_amdgcn_wmma_f32_16x16x4_f32(false, a1, false, bones,
                                                     (short)0, csum, false, false);
        csq  = __builtin_amdgcn_wmma_f32_16x16x4_f32(false, q1, false, bones,
                                                     (short)0, csq,  false, false);
    }
    if (it < nfull) {                    // odd full chunk
        v2f a;
        a.x = px[0];
        a.y = px[FEAT];
        v2f q = a * a;
        csum = __builtin_amdgcn_wmma_f32_16x16x4_f32(false, a, false, bones,
                                                     (short)0, csum, false, false);
        csq  = __builtin_amdgcn_wmma_f32_16x16x4_f32(false, q, false, bones,
                                                     (short)0, csq,  false, false);
    }
    {                                    // tail: remaining c & 3 rows, guarded
        const int r0 = s + (nfull << 2);
        if (r0 < e) {
            const int ra = r0 + rx;
            const int rb = ra + 1;
            v2f a;
            a.x = (ra < e) ? x[(size_t)ra * FEAT + feat] : 0.0f;
            a.y = (rb < e) ? x[(size_t)rb * FEAT + feat] : 0.0f;
            v2f q = a * a;
            csum = __builtin_amdgcn_wmma_f32_16x16x4_f32(false, a, false, bones,
                                                         (short)0, csum, false, false);
            csq  = __builtin_amdgcn_wmma_f32_16x16x4_f32(false, q, false, bones,
                                                         (short)0, csq,  false, false);
        }
    }

    // C/D layout: VGPR r holds M=r (lanes 0-15) and M=8+r (lanes 16-31);
    // every N column equals the row-sum, so one lane per half suffices.
    if (lane == 0) {
        #pragma unroll
        for (int m = 0; m < 8; ++m) {
            lds_a[f0 + m] = csum[m];
            lds_b[f0 + m] = csq[m];
        }
    }
    if (lane == 16) {
        #pragma unroll
        for (int m = 0; m < 8; ++m) {
            lds_a[f0 + 8 + m] = csum[m];
            lds_b[f0 + 8 + m] = csq[m];
        }
    }
    __syncthreads();

    // ---------------- mean / inv-std (ddof = 1) ----------------
    if (tid < FEAT) {
        const float cnt  = (float)c;
        const float sum  = lds_a[tid];
        const float ss   = lds_b[tid];
        const float mean = sum / cnt;
        const float var  = (ss - cnt * mean * mean) / fmaxf(cnt - 1.0f, 1.0f);
        const float stdv = sqrtf(fmaxf(var, 0.0f));
        lds_a[tid] = mean;                 // same-thread same-slot rewrite: safe
        lds_b[tid] = 1.0f / (stdv + EPS);
    }
    __syncthreads();

    // ---------------- Pass B: normalize, float4 streaming ----------------
    // eidx = tid*4 + it*2048; 2048 % 256 == 0 -> feature slot fb is
    // loop-invariant, so all per-feature constants hoist out of the loop.
    const int fb = (tid << 2) & (FEAT - 1);
    const f4 gv = *(const f4*)(gamma + fb);
    const f4 bv = *(const f4*)(beta  + fb);
    f4 mv, iv;
    mv.x = lds_a[fb + 0]; mv.y = lds_a[fb + 1]; mv.z = lds_a[fb + 2]; mv.w = lds_a[fb + 3];
    iv.x = lds_b[fb + 0]; iv.y = lds_b[fb + 1]; iv.z = lds_b[fb + 2]; iv.w = lds_b[fb + 3];

    const size_t total = (size_t)c * FEAT;
    for (size_t eidx = (size_t)tid * 4; eidx < total; eidx += 512 * 4) {
        const size_t row = (size_t)s + (eidx >> 8);
        const size_t off = row * FEAT + fb;
        __builtin_prefetch(x + off + 8 * FEAT, 0, 3);
        const f4 xv = *(const f4*)(x + off);
        f4 ov;
        ov.x = gv.x * ((xv.x - mv.x) * iv.x) + bv.x;
        ov.y = gv.y * ((xv.y - mv.y) * iv.y) + bv.y;
        ov.z = gv.z * ((xv.z - mv.z) * iv.z) + bv.z;
        ov.w = gv.w * ((xv.w - mv.w) * iv.w) + bv.w;
        __builtin_nontemporal_store(ov, (f4*)(out + off));  // write-once stream
    }
}

// ---------------------------------------------------------------------------
extern "C" void kernel_launch(void* const* d_in, const int* in_sizes, int n_in,
                              void* d_out, int out_size, void* d_ws, size_t ws_size,
                              hipStream_t stream) {
    const float* x      = (const float*)d_in[0];
    const float* gamma  = (const float*)d_in[1];
    const float* beta   = (const float*)d_in[2];
    const int*   segids = (const int*)d_in[3];
    const int n_nodes   = in_sizes[3];

    int* starts = (int*)d_ws;  // NGRAPH + 1 ints

    seg_starts_kernel<<<(NGRAPH + 1 + 255) / 256, 256, 0, stream>>>(segids, starts, n_nodes);
    graphnorm_kernel<<<NGRAPH, 512, 0, stream>>>(x, gamma, beta, starts, (float*)d_out);
}